// BCGNConv_26998164422989
// MI455X (gfx1250) — compile-verified
//
#include <hip/hip_runtime.h>

typedef float v2f __attribute__((ext_vector_type(2)));
typedef float v8f __attribute__((ext_vector_type(8)));

#define IN_F 256
#define OUT_F 64
#define LDS_PITCH 260   // 256 + 4 pad floats: distinct banks across lanes, keeps 16B align

// ---------------- zero output + degree arrays ----------------
__global__ void bcg_zero(float* __restrict__ out, int n_out,
                         float* __restrict__ deg, int n_deg) {
    int i = blockIdx.x * blockDim.x + threadIdx.x;
    int stride = gridDim.x * blockDim.x;
    for (int j = i; j < n_out; j += stride) out[j] = 0.0f;
    for (int j = i; j < n_deg; j += stride) deg[j] = 0.0f;
}

// ---------------- degree counts (f32 global atomics) ----------------
__global__ void bcg_degrees(const int* __restrict__ src, const int* __restrict__ dst,
                            float* __restrict__ out_deg, float* __restrict__ in_deg, int E) {
    int e = blockIdx.x * blockDim.x + threadIdx.x;
    if (e < E) {
        unsafeAtomicAdd(&out_deg[src[e]], 1.0f);
        unsafeAtomicAdd(&in_deg[dst[e]], 1.0f);
    }
}

// ---------------- Bayesian sampling + KL; weight written in WMMA B-fragment order ----
// Packed layout: element (k,n) -> wpack[(k>>2)*256 + ((k>>1)&1)*128 + n*2 + (k&1)]
// so a wave's B fragment {K=khalf, K=khalf+1} at column n is one aligned b64.
__global__ void bcg_sample_kl(const float* __restrict__ wmu, const float* __restrict__ wls,
                              const float* __restrict__ wpm, const float* __restrict__ wpl,
                              const float* __restrict__ bmu, const float* __restrict__ bls,
                              const float* __restrict__ bpm, const float* __restrict__ bpl,
                              const float* __restrict__ eps_w, const float* __restrict__ eps_b,
                              float* __restrict__ wpack, float* __restrict__ bias,
                              float* __restrict__ kl_out, int n_w, int n_b) {
    __shared__ float red[256];
    int i = blockIdx.x * 256 + threadIdx.x;
    float kl = 0.0f;
    if (i < n_w) {
        float mu = wmu[i], ls = wls[i], pm = wpm[i], pl = wpl[i];
        int k = i >> 6, n = i & 63;
        int dest = (k >> 2) * 256 + ((k >> 1) & 1) * 128 + n * 2 + (k & 1);
        wpack[dest] = mu + expf(ls) * eps_w[i];
        float var_q = expf(2.0f * ls), var_p = expf(2.0f * pl);
        float d = mu - pm;
        kl = pl - ls + (var_q + d * d) / (2.0f * var_p) - 0.5f;
    } else if (i < n_w + n_b) {
        int j = i - n_w;
        float mu = bmu[j], ls = bls[j], pm = bpm[j], pl = bpl[j];
        bias[j] = mu + expf(ls) * eps_b[j];
        float var_q = expf(2.0f * ls), var_p = expf(2.0f * pl);
        float d = mu - pm;
        kl = pl - ls + (var_q + d * d) / (2.0f * var_p) - 0.5f;
    }
    red[threadIdx.x] = kl;
    __syncthreads();
    for (int s = 128; s > 0; s >>= 1) {
        if (threadIdx.x < s) red[threadIdx.x] += red[threadIdx.x + s];
        __syncthreads();
    }
    if (threadIdx.x == 0) unsafeAtomicAdd(kl_out, red[0]);
}

// ---------------- fp32 WMMA GEMM: h = (feat * out_deg^-1/2) @ weight ----------------
// block = 128 threads (4 wave32), one 16-row stripe per block; wave w -> cols [16w,16w+16)
__global__ void bcg_gemm_wmma(const float* __restrict__ feat,
                              const float* __restrict__ out_deg,
                              const float* __restrict__ wpack,
                              float* __restrict__ h, int N) {
    __shared__ float lds[16 * LDS_PITCH];
    const int block_row = blockIdx.x * 16;
    const int tid  = threadIdx.x;
    const int wave = tid >> 5;
    const int lane = tid & 31;
    const bool full = (block_row + 16 <= N);

    // Cooperative staged load: 16 rows x 256 floats, scaled by left norm, float4 moves
    if (full) {
        for (int i = tid; i < 16 * (IN_F / 4); i += 128) {
            int r  = i >> 6;
            int c4 = i & 63;
            float nl = rsqrtf(fmaxf(out_deg[block_row + r], 1.0f));
            float4 f = ((const float4*)(feat + (size_t)(block_row + r) * IN_F))[c4];
            ((float4*)(lds + r * LDS_PITCH))[c4] =
                make_float4(f.x * nl, f.y * nl, f.z * nl, f.w * nl);
        }
    } else {
        for (int i = tid; i < 16 * (IN_F / 4); i += 128) {
            int r  = i >> 6;
            int c4 = i & 63;
            float4 v = make_float4(0.f, 0.f, 0.f, 0.f);
            int row = block_row + r;
            if (row < N) {
                float nl = rsqrtf(fmaxf(out_deg[row], 1.0f));
                float4 f = ((const float4*)(feat + (size_t)row * IN_F))[c4];
                v = make_float4(f.x * nl, f.y * nl, f.z * nl, f.w * nl);
            }
            ((float4*)(lds + r * LDS_PITCH))[c4] = v;
        }
    }
    __syncthreads();

    const int sub   = lane & 15;          // M (for A) / N (for B,D) index within tile
    const int hi    = lane >> 4;          // 0: K=0,1 ; 1: K=2,3
    const int khalf = hi * 2;
    const int n     = wave * 16 + sub;    // global output column 0..63
    const float* bptr = wpack + hi * 128 + (size_t)n * 2;   // + (k0>>2)*256 per step

    v8f acc = {0.f, 0.f, 0.f, 0.f, 0.f, 0.f, 0.f, 0.f};
    #pragma unroll 8
    for (int k0 = 0; k0 < IN_F; k0 += 4) {
        // A fragment (16x4 fp32): lane holds {K=khalf, K=khalf+1} of row M=sub
        v2f a = *(const v2f*)(lds + sub * LDS_PITCH + k0 + khalf);
        // B fragment: one aligned b64 from the pre-swizzled weight
        v2f b = *(const v2f*)(bptr + (k0 >> 2) * 256);
        acc = __builtin_amdgcn_wmma_f32_16x16x4_f32(
                  false, a, false, b, (short)0, acc, false, false);
    }

    // D layout: VGPR r -> M = r + 8*hi, N = lane%16
    const int mbase = hi * 8;
    if (full) {
        float* hp = h + (size_t)(block_row + mbase) * OUT_F + n;
        #pragma unroll
        for (int r = 0; r < 8; ++r) hp[(size_t)r * OUT_F] = acc[r];
    } else {
        #pragma unroll
        for (int r = 0; r < 8; ++r) {
            int row = block_row + mbase + r;
            if (row < N) h[(size_t)row * OUT_F + n] = acc[r];
        }
    }
}

// ---------------- edge gather + scatter-add (L2-resident atomics) ----------------
__global__ void bcg_scatter(const float* __restrict__ h,
                            const int* __restrict__ src, const int* __restrict__ dst,
                            float* __restrict__ out, int total) {
    int i = blockIdx.x * blockDim.x + threadIdx.x;
    if (i < total) {
        int e = i >> 6;
        int f = i & 63;
        int s = src[e], d = dst[e];
        unsafeAtomicAdd(&out[(size_t)d * OUT_F + f], h[(size_t)s * OUT_F + f]);
    }
}

// ---------------- right norm + bias ----------------
__global__ void bcg_finalize(float* __restrict__ out, const float* __restrict__ in_deg,
                             const float* __restrict__ bias, int total) {
    int i = blockIdx.x * blockDim.x + threadIdx.x;
    if (i < total) {
        int n = i >> 6;
        int f = i & 63;
        out[i] = out[i] * rsqrtf(fmaxf(in_deg[n], 1.0f)) + bias[f];
    }
}

extern "C" void kernel_launch(void* const* d_in, const int* in_sizes, int n_in,
                              void* d_out, int out_size, void* d_ws, size_t ws_size,
                              hipStream_t stream) {
    const float* feat  = (const float*)d_in[0];
    const float* wmu   = (const float*)d_in[1];
    const float* wls   = (const float*)d_in[2];
    const float* wpm   = (const float*)d_in[3];
    const float* wpl   = (const float*)d_in[4];
    const float* bmu   = (const float*)d_in[5];
    const float* bls   = (const float*)d_in[6];
    const float* bpm   = (const float*)d_in[7];
    const float* bpl   = (const float*)d_in[8];
    const float* eps_w = (const float*)d_in[9];
    const float* eps_b = (const float*)d_in[10];
    const int*   src   = (const int*)d_in[11];
    const int*   dst   = (const int*)d_in[12];

    const int N = in_sizes[0] / IN_F;
    const int E = in_sizes[11];

    float* out = (float*)d_out;
    float* ws  = (float*)d_ws;
    float* out_deg = ws;                    // N
    float* in_deg  = ws + N;                // N
    float* wpack   = ws + 2 * (size_t)N;    // 256*64 (fragment-swizzled sampled weight)
    float* bias    = wpack + IN_F * OUT_F;  // 64
    float* h       = bias + OUT_F;          // N*64

    const int n_out_elems = N * OUT_F + 1;  // rst + kl scalar
    const int n_w = IN_F * OUT_F, n_b = OUT_F;

    bcg_zero<<<(n_out_elems + 255) / 256, 256, 0, stream>>>(out, n_out_elems, out_deg, 2 * N);
    bcg_degrees<<<(E + 255) / 256, 256, 0, stream>>>(src, dst, out_deg, in_deg, E);
    bcg_sample_kl<<<(n_w + n_b + 255) / 256, 256, 0, stream>>>(
        wmu, wls, wpm, wpl, bmu, bls, bpm, bpl, eps_w, eps_b,
        wpack, bias, out + (size_t)N * OUT_F, n_w, n_b);
    bcg_gemm_wmma<<<(N + 15) / 16, 128, 0, stream>>>(feat, out_deg, wpack, h, N);
    const long long total_sc = (long long)E * OUT_F;
    bcg_scatter<<<(int)((total_sc + 255) / 256), 256, 0, stream>>>(h, src, dst, out, (int)total_sc);
    bcg_finalize<<<(N * OUT_F + 255) / 256, 256, 0, stream>>>(out, in_deg, bias, N * OUT_F);
}